// GraphEncoder_9912784519798
// MI455X (gfx1250) — compile-verified
//
#include <hip/hip_runtime.h>
#include <math.h>

#define D 128
#define G 256
#define EPS 1e-5f

typedef float v2f __attribute__((ext_vector_type(2)));
typedef float v8f __attribute__((ext_vector_type(8)));

// ---------------------------------------------------------------- elementwise

__global__ void gather_kernel(const int* __restrict__ idx,
                              const float* __restrict__ emb,
                              float* __restrict__ out, int N) {
    long t = (long)blockIdx.x * blockDim.x + threadIdx.x;
    int n = (int)(t >> 5), q = (int)(t & 31);          // 32 lanes * float4 = 128
    if (n >= N) return;
    const float4* src = (const float4*)(emb + (long)idx[n] * D);
    float4* dst = (float4*)(out + (long)n * D);
    dst[q] = src[q];
}

__global__ void deg_kernel(const int* __restrict__ ei, const float* __restrict__ ew,
                           float* __restrict__ deg, int E) {
    int e = blockIdx.x * blockDim.x + threadIdx.x;
    if (e >= E) return;
    unsafeAtomicAdd(&deg[ei[E + e]], ew[e]);           // col = ei[1][e]
}

__global__ void dis_kernel(float* __restrict__ deg, int N) {
    int n = blockIdx.x * blockDim.x + threadIdx.x;
    if (n >= N) return;
    float d = deg[n];
    deg[n] = d > 0.f ? rsqrtf(fmaxf(d, EPS)) : 0.f;
}

__global__ void norm_kernel(const int* __restrict__ ei, const float* __restrict__ ew,
                            const float* __restrict__ dis, float* __restrict__ nrm, int E) {
    int e = blockIdx.x * blockDim.x + threadIdx.x;
    if (e >= E) return;
    nrm[e] = dis[ei[e]] * ew[e] * dis[ei[E + e]];
}

// ---------------------------------------------------------------- batch norm

__global__ void bn_stats_kernel(const float* __restrict__ x, float* __restrict__ stats, int N) {
    int d = threadIdx.x;                               // 128 threads = features
    float s = 0.f, s2 = 0.f;
    for (int n = blockIdx.x; n < N; n += gridDim.x) {
        float v = x[(long)n * D + d];
        s += v; s2 += v * v;
    }
    unsafeAtomicAdd(&stats[d], s);
    unsafeAtomicAdd(&stats[D + d], s2);
}

__global__ void bn_final_kernel(const float* __restrict__ stats,
                                const float* __restrict__ gamma,
                                const float* __restrict__ beta,
                                float* __restrict__ scsh, float invN) {
    int d = threadIdx.x;
    float mean = stats[d] * invN;
    float var  = stats[D + d] * invN - mean * mean;    // biased var
    float sc   = rsqrtf(var + EPS) * gamma[d];
    scsh[d]     = sc;
    scsh[D + d] = beta[d] - mean * sc;
}

__global__ void bn_apply_kernel(float* __restrict__ x, const float* __restrict__ scsh, long total4) {
    long t = (long)blockIdx.x * blockDim.x + threadIdx.x;
    if (t >= total4) return;
    int d0 = (int)(t & 31) * 4;
    float4 v = ((float4*)x)[t];
    v.x = v.x * scsh[d0 + 0] + scsh[D + d0 + 0];
    v.y = v.y * scsh[d0 + 1] + scsh[D + d0 + 1];
    v.z = v.z * scsh[d0 + 2] + scsh[D + d0 + 2];
    v.w = v.w * scsh[d0 + 3] + scsh[D + d0 + 3];
    ((float4*)x)[t] = v;
}

// ---------------------------------------------------------------- WMMA GEMM: H = A(NxD) * W(DxD)
//
// W staged in LDS K-pair interleaved: Wlds[(k>>1)*256 + n*2 + (k&1)] = W[k][n]
// so each lane's B fragment {W[k][n], W[k+1][n]} is one aligned ds_load_b64
// straight into the even VGPR pair WMMA wants (no v_mov shuffling).

__global__ void __launch_bounds__(256, 1)
gemm_kernel(const float* __restrict__ Amat, const float* __restrict__ W,
            float* __restrict__ Hmat, int N) {
    __shared__ float Wlds[D * D];                      // 64 KB of the 320 KB WGP LDS
    int tid = threadIdx.x;
    for (int i = tid; i < D * D; i += 256) {
        int k = i >> 7, n = i & 127;
        Wlds[((k >> 1) << 8) + (n << 1) + (k & 1)] = W[i];
    }
    __syncthreads();

    int wave = tid >> 5;
    int lane = tid & 31;
    int rowBase = blockIdx.x * 128 + wave * 16;        // wave-uniform (EXEC all-ones in WMMA)
    if (rowBase >= N) return;                          // N % 16 == 0 -> whole-wave guard

    int laneM = lane & 15;                             // M index for A, N index for B/C/D
    int laneK = (lane >> 4) << 1;                      // K sub-offset 0 or 2 (always even)

    v8f acc[8];
    v8f zero = {0.f, 0.f, 0.f, 0.f, 0.f, 0.f, 0.f, 0.f};
#pragma unroll
    for (int i = 0; i < 8; ++i) acc[i] = zero;

    const float* aRow = Amat + (long)(rowBase + laneM) * D + laneK;
    const float* bCol = Wlds + ((laneK >> 1) << 8) + (laneM << 1);

    for (int kk = 0; kk < D; kk += 4) {
        v2f a = *(const v2f*)(aRow + kk);              // A 16x4 frag (one global_load_b64)
        const float* bk = bCol + ((kk >> 1) << 8);
#pragma unroll
        for (int nt = 0; nt < 8; ++nt) {
            v2f b = *(const v2f*)(bk + (nt << 5));     // B 4x16 frag (one ds_load_b64)
            acc[nt] = __builtin_amdgcn_wmma_f32_16x16x4_f32(
                false, a, false, b, (short)0, acc[nt], false, false);
        }
    }

    int rowOff = (lane >> 4) * 8;                      // D layout: M=j (lanes<16), M=8+j (lanes>=16)
#pragma unroll
    for (int nt = 0; nt < 8; ++nt) {
#pragma unroll
        for (int j = 0; j < 8; ++j) {
            Hmat[(long)(rowBase + rowOff + j) * D + nt * 16 + laneM] = acc[nt][j];
        }
    }
}

// ---------------------------------------------------------------- edge scatter (the bandwidth hog)

__global__ void scatter_kernel(const int* __restrict__ ei, const float* __restrict__ nrm,
                               const float* __restrict__ H, float* __restrict__ Out, int E) {
    long t = (long)blockIdx.x * blockDim.x + threadIdx.x;
    int e = (int)(t >> 5);
    if (e >= E) return;
    int lane = (int)(t & 31);
    float w = nrm[e];
    int r = ei[e], c = ei[E + e];
    float4 hv = ((const float4*)(H + (long)r * D))[lane];
    float* o = Out + (long)c * D + lane * 4;
    unsafeAtomicAdd(o + 0, w * hv.x);
    unsafeAtomicAdd(o + 1, w * hv.y);
    unsafeAtomicAdd(o + 2, w * hv.z);
    unsafeAtomicAdd(o + 3, w * hv.w);
}

__global__ void bias_relu_kernel(float* __restrict__ x, const float* __restrict__ b, long total4) {
    long t = (long)blockIdx.x * blockDim.x + threadIdx.x;
    if (t >= total4) return;
    int d0 = (int)(t & 31) * 4;
    float4 v = ((float4*)x)[t];
    v.x = fmaxf(v.x + b[d0 + 0], 0.f);
    v.y = fmaxf(v.y + b[d0 + 1], 0.f);
    v.z = fmaxf(v.z + b[d0 + 2], 0.f);
    v.w = fmaxf(v.w + b[d0 + 3], 0.f);
    ((float4*)x)[t] = v;
}

// ---------------------------------------------------------------- segment softmax + pooling

__global__ void seg_max_kernel(const float* __restrict__ tf, const int* __restrict__ batch,
                               float* __restrict__ m, int N) {
    int n = blockIdx.x * blockDim.x + threadIdx.x;
    if (n >= N) return;
    // tfidf >= 0, so int-bit ordering == float ordering; m pre-zeroed (0.0f)
    atomicMax((int*)&m[batch[n]], __float_as_int(tf[n]));
}

__global__ void seg_exp_kernel(const float* __restrict__ tf, const int* __restrict__ batch,
                               const float* __restrict__ m, float* __restrict__ coef,
                               float* __restrict__ s, int N) {
    int n = blockIdx.x * blockDim.x + threadIdx.x;
    if (n >= N) return;
    int g = batch[n];
    float e = __expf(tf[n] - m[g]);
    coef[n] = e;
    unsafeAtomicAdd(&s[g], e);
}

__global__ void seg_div_kernel(float* __restrict__ coef, const int* __restrict__ batch,
                               const float* __restrict__ s, int N) {
    int n = blockIdx.x * blockDim.x + threadIdx.x;
    if (n >= N) return;
    coef[n] /= s[batch[n]];
}

__global__ void pool_kernel(const float* __restrict__ coef, const int* __restrict__ batch,
                            const float* __restrict__ x, float* __restrict__ out, int N) {
    long t = (long)blockIdx.x * blockDim.x + threadIdx.x;
    int n = (int)(t >> 5);
    if (n >= N) return;
    int lane = (int)(t & 31);
    float w = coef[n];
    int g = batch[n];
    float4 v = ((const float4*)(x + (long)n * D))[lane];
    float* o = out + (long)g * D + lane * 4;
    unsafeAtomicAdd(o + 0, w * v.x);
    unsafeAtomicAdd(o + 1, w * v.y);
    unsafeAtomicAdd(o + 2, w * v.z);
    unsafeAtomicAdd(o + 3, w * v.w);
}

// ---------------------------------------------------------------- host side

static inline size_t alignup(size_t x) { return (x + 255) & ~(size_t)255; }

extern "C" void kernel_launch(void* const* d_in, const int* in_sizes, int n_in,
                              void* d_out, int out_size, void* d_ws, size_t ws_size,
                              hipStream_t stream) {
    const int*   x_index = (const int*)  d_in[0];
    const float* tfidf   = (const float*)d_in[1];
    const int*   ei      = (const int*)  d_in[2];   // [2, E] flattened
    const int*   batch   = (const int*)  d_in[3];
    const float* eattr   = (const float*)d_in[4];
    const float* emb     = (const float*)d_in[5];
    const float* gamma1  = (const float*)d_in[6];
    const float* beta1   = (const float*)d_in[7];
    const float* W1      = (const float*)d_in[8];
    const float* b1      = (const float*)d_in[9];
    const float* gamma2  = (const float*)d_in[10];
    const float* beta2   = (const float*)d_in[11];
    const float* W2      = (const float*)d_in[12];
    const float* b2      = (const float*)d_in[13];

    const int N = in_sizes[0];
    const int E = in_sizes[4];

    char* ws = (char*)d_ws;
    size_t NB = (size_t)N * D * sizeof(float);
    size_t offA     = 0;
    size_t offB     = alignup(offA + NB);
    size_t offDis   = alignup(offB + NB);
    size_t offNorm  = alignup(offDis + (size_t)N * 4);
    size_t offStats = alignup(offNorm + (size_t)E * 4);
    size_t offScsh  = alignup(offStats + 2 * D * 4);
    size_t offM     = alignup(offScsh + 2 * D * 4);
    size_t offS     = alignup(offM + G * 4);
    size_t offCoef  = alignup(offS + G * 4);

    float* A     = (float*)(ws + offA);
    float* B     = (float*)(ws + offB);
    float* dis   = (float*)(ws + offDis);
    float* nrm   = (float*)(ws + offNorm);
    float* stats = (float*)(ws + offStats);
    float* scsh  = (float*)(ws + offScsh);
    float* m     = (float*)(ws + offM);
    float* s     = (float*)(ws + offS);
    float* coef  = (float*)(ws + offCoef);
    float* out   = (float*)d_out;

    const int T = 256;
    long nd4 = (long)N * 32;                       // N*D/4 float4 slots
    int gN    = (N + T - 1) / T;
    int gE    = (E + T - 1) / T;
    int gND4  = (int)((nd4 + T - 1) / T);
    int gE32  = (int)(((long)E * 32 + T - 1) / T);
    int gGemm = (N + 127) / 128;
    float invN = 1.0f / (float)N;

    // Stage 0: embedding gather; edge normalization factors (shared by both layers)
    gather_kernel<<<gND4, T, 0, stream>>>(x_index, emb, A, N);
    hipMemsetAsync(dis, 0, (size_t)N * 4, stream);
    deg_kernel<<<gE, T, 0, stream>>>(ei, eattr, dis, E);
    dis_kernel<<<gN, T, 0, stream>>>(dis, N);
    norm_kernel<<<gE, T, 0, stream>>>(ei, eattr, dis, nrm, E);

    // Layer 1: BN -> GEMM(WMMA) -> scatter -> bias+ReLU
    hipMemsetAsync(stats, 0, 2 * D * 4, stream);
    bn_stats_kernel<<<512, D, 0, stream>>>(A, stats, N);
    bn_final_kernel<<<1, D, 0, stream>>>(stats, gamma1, beta1, scsh, invN);
    bn_apply_kernel<<<gND4, T, 0, stream>>>(A, scsh, nd4);
    gemm_kernel<<<gGemm, T, 0, stream>>>(A, W1, B, N);
    hipMemsetAsync(A, 0, NB, stream);
    scatter_kernel<<<gE32, T, 0, stream>>>(ei, nrm, B, A, E);
    bias_relu_kernel<<<gND4, T, 0, stream>>>(A, b1, nd4);

    // Layer 2
    hipMemsetAsync(stats, 0, 2 * D * 4, stream);
    bn_stats_kernel<<<512, D, 0, stream>>>(A, stats, N);
    bn_final_kernel<<<1, D, 0, stream>>>(stats, gamma2, beta2, scsh, invN);
    bn_apply_kernel<<<gND4, T, 0, stream>>>(A, scsh, nd4);
    gemm_kernel<<<gGemm, T, 0, stream>>>(A, W2, B, N);
    hipMemsetAsync(A, 0, NB, stream);
    scatter_kernel<<<gE32, T, 0, stream>>>(ei, nrm, B, A, E);
    bias_relu_kernel<<<gND4, T, 0, stream>>>(A, b2, nd4);

    // Pooling: segment softmax over tfidf, weighted segment sum
    hipMemsetAsync(m, 0, G * 4, stream);
    hipMemsetAsync(s, 0, G * 4, stream);
    hipMemsetAsync(out, 0, (size_t)G * D * 4, stream);
    seg_max_kernel<<<gN, T, 0, stream>>>(tfidf, batch, m, N);
    seg_exp_kernel<<<gN, T, 0, stream>>>(tfidf, batch, m, coef, s, N);
    seg_div_kernel<<<gN, T, 0, stream>>>(coef, batch, s, N);
    pool_kernel<<<gND4, T, 0, stream>>>(coef, batch, A, out, N);
}